// Embedding_26465588478015
// MI455X (gfx1250) — compile-verified
//
#include <hip/hip_runtime.h>

// CDNA5 / gfx1250: wave32, WMMA (not MFMA).
// out[b,s,e] = sum_i in[b,i,s] * emb[i,e]   (B=64, I=K=64, S=4096, E=64, fp32)
//
// Roofline: 2.15 GFLOP vs 134 MB HBM traffic -> 5.8 us memory floor @ 23.3 TB/s;
// f32 WMMA (~1/32 of fp8 dense matrix rate) lands ~6.8 us -> near-balanced, so we
// keep exact fp32 semantics and focus on removing VALU/DS overhead around WMMA.

typedef float v2f __attribute__((ext_vector_type(2)));
typedef float v8f __attribute__((ext_vector_type(8)));

#define BATCH     64
#define SEQ       4096
#define KDIM      64
#define EDIM      64
#define SBLOCK    128                      // s-rows per workgroup (8 waves x 16)
#define BLOCKS_PER_BATCH (SEQ / SBLOCK)    // 32

// LDS layout for embedding: fragment-interleaved so each B fragment (two
// consecutive K values, same e) is one aligned 8-byte LDS word:
//   lds[(i>>1)*128 + 2*e + (i&1)] = emb[i*64 + e]
__global__ __launch_bounds__(256, 2)
void embed_gemm_wmma_f32(const float* __restrict__ in,   // [B, I, S]
                         const float* __restrict__ emb,  // [I, E]
                         float* __restrict__ out)        // [B, S, E]
{
    __shared__ float lds_emb[KDIM * EDIM];   // 16 KB of 320 KB/WGP LDS

    // ---- stage embedding into LDS, permuted (float4 global reads) ----
    {
        const float4* src = reinterpret_cast<const float4*>(emb);
        const int t = threadIdx.x;
#pragma unroll
        for (int q = 0; q < 4; ++q) {
            const int f   = t + 256 * q;        // float4 index 0..1023
            const float4 v = src[f];
            const int idx = f * 4;              // flat element index
            const int i   = idx >> 6;           // row (K)
            const int e   = idx & 63;           // col (E), multiple of 4
            float* dstp = &lds_emb[(i >> 1) * 128 + (i & 1) + 2 * e];
            dstp[0] = v.x; dstp[2] = v.y; dstp[4] = v.z; dstp[6] = v.w;
        }
    }
    __syncthreads();

    const int wave  = threadIdx.x >> 5;   // 0..7
    const int lane  = threadIdx.x & 31;
    const int m     = lane & 15;          // M (=s) / N (=e) index within fragment
    const int khalf = lane >> 4;          // 0: K 0..1 , 1: K 2..3  (ISA 16x4 layout)

    const int b    = blockIdx.x >> 5;     // / BLOCKS_PER_BATCH
    const int sblk = blockIdx.x & 31;
    const int s0   = sblk * SBLOCK + wave * 16;   // this wave's first s-row

    const float* inb = in + (size_t)b * KDIM * SEQ;  // in[b, i, s]

    v8f acc0 = {}, acc1 = {}, acc2 = {}, acc3 = {};

#pragma unroll
    for (int k = 0; k < KDIM / 4; ++k) {
        const int i0 = 4 * k + 2 * khalf;   // K index of A/B vgpr0 for this lane

        // A fragment: A[m = s-row, K = i]; vgpr j holds K = i0 + j (coalesced 64B/half-wave)
        v2f a;
        a.x = inb[(size_t)(i0 + 0) * SEQ + s0 + m];
        a.y = inb[(size_t)(i0 + 1) * SEQ + s0 + m];

        // B fragments: one aligned b64 LDS load each (no register shuffling)
        const v2f* brow = reinterpret_cast<const v2f*>(
            &lds_emb[(2 * k + khalf) * 128 + 2 * m]);
        const v2f b0 = brow[0];    // e = m
        const v2f b1 = brow[16];   // e = m + 16
        const v2f b2 = brow[32];   // e = m + 32
        const v2f b3 = brow[48];   // e = m + 48

        // D = A x B + C   (fp32 WMMA, exact vs the fp32 reference)
        acc0 = __builtin_amdgcn_wmma_f32_16x16x4_f32(false, a, false, b0, (short)0, acc0, false, false);
        acc1 = __builtin_amdgcn_wmma_f32_16x16x4_f32(false, a, false, b1, (short)0, acc1, false, false);
        acc2 = __builtin_amdgcn_wmma_f32_16x16x4_f32(false, a, false, b2, (short)0, acc2, false, false);
        acc3 = __builtin_amdgcn_wmma_f32_16x16x4_f32(false, a, false, b3, (short)0, acc3, false, false);
    }

    // ---- store: C/D layout -> VGPR r, lane: (M = r + 8*khalf, N = m) ----
    float* outp = out + ((size_t)b * SEQ + s0) * EDIM;
#pragma unroll
    for (int r = 0; r < 8; ++r) {
        float* rowp = outp + (size_t)(r + 8 * khalf) * EDIM + m;
        rowp[ 0] = acc0[r];
        rowp[16] = acc1[r];
        rowp[32] = acc2[r];
        rowp[48] = acc3[r];
    }
}

extern "C" void kernel_launch(void* const* d_in, const int* in_sizes, int n_in,
                              void* d_out, int out_size, void* d_ws, size_t ws_size,
                              hipStream_t stream) {
    const float* in  = (const float*)d_in[0];   // [64, 64, 4096]
    const float* emb = (const float*)d_in[1];   // [64, 64]
    float*       out = (float*)d_out;           // [64, 4096, 64]

    dim3 grid(BATCH * BLOCKS_PER_BATCH);  // 2048
    dim3 block(256);                      // 8 waves (wave32)
    hipLaunchKernelGGL(embed_gemm_wmma_f32, grid, block, 0, stream, in, emb, out);
}